// PhysGATModel_63625645523025
// MI455X (gfx1250) — compile-verified
//
#include <hip/hip_runtime.h>
#include <math.h>

typedef __attribute__((ext_vector_type(2))) float v2f;
typedef __attribute__((ext_vector_type(8))) float v8f;

#define NR 20000
#define NS 5000
#define ERR 320000
#define ERS 200000

// ---------- helpers: monotone uint encoding of float for atomic max ----------
__device__ __forceinline__ unsigned f2ord(float f) {
  unsigned u = __float_as_uint(f);
  return (u & 0x80000000u) ? ~u : (u | 0x80000000u);
}
__device__ __forceinline__ float ord2f(unsigned u) {
  return (u & 0x80000000u) ? __uint_as_float(u & 0x7fffffffu)
                           : __uint_as_float(~u);
}

__global__ void fill_f32(float* p, float v, int n) {
  int i = blockIdx.x * blockDim.x + threadIdx.x;
  if (i < n) p[i] = v;
}
__global__ void fill_u32(unsigned* p, unsigned v, int n) {
  int i = blockIdx.x * blockDim.x + threadIdx.x;
  if (i < n) p[i] = v;
}

// ---------------- fp32 GEMM on matrix pipes: V_WMMA_F32_16X16X4_F32 ----------
// C[M,N](ldc) = A[M,K](lda) @ W[K,N] + b1 + b2 ; optional leaky-relu(slope)
// block = 128 threads (4 waves) -> 64 rows x 64 cols tile.
// B staged in LDS as a 32(K) x 64(N) tile, padded stride 72 so the two
// lane-halves of a wave hit disjoint bank groups. Per K-chunk: 8 steps x
// 4 column groups = 32 WMMAs, A = one masked global b64 load per step.
#define BK 32
#define BS 72   // padded LDS row stride (floats)

__global__ void gemm_bias_act(const float* __restrict__ A, int lda,
                              const float* __restrict__ W,
                              const float* __restrict__ b1,
                              const float* __restrict__ b2,
                              float* __restrict__ C, int ldc,
                              int M, int N, int K, float slope) {
  __shared__ float Bs[BK * BS];

  const int tid   = threadIdx.x;
  const int lane  = tid & 31;
  const int wave  = tid >> 5;
  const int l16   = lane & 15;
  const int lhalf = lane >> 4;                 // 0: K 0..1 | 1: K 2..3
  const int row0  = blockIdx.y * 64 + wave * 16;
  const int colb  = blockIdx.x * 64;           // block column base
  const int mr    = row0 + l16;
  const bool mok  = (mr < M);
  const float msk = mok ? 1.f : 0.f;
  const float* Arow = A + (long)(mok ? mr : 0) * lda;  // clamped: loads legal

  v8f acc[4] = {{}, {}, {}, {}};

  for (int k0 = 0; k0 < K; k0 += BK) {
    // cooperative load of B[k0..k0+31][colb..colb+63] into LDS (16 elem/thread)
#pragma unroll
    for (int i = 0; i < 16; ++i) {
      int idx = i * 128 + tid;
      int r = idx >> 6, c = idx & 63;
      Bs[r * BS + c] = W[(long)(k0 + r) * N + colb + c];
      if (k0 + BK < K)
        __builtin_prefetch(W + (long)(k0 + BK + r) * N + colb + c, 0, 0);
    }
    __syncthreads();

#pragma unroll
    for (int ks = 0; ks < 8; ++ks) {
      const int ka = ks * 4 + lhalf * 2;       // local K pair for this half
      v2f a = *(const v2f*)(Arow + k0 + ka);
      a.x *= msk;
      a.y *= msk;
      const float* br0 = Bs + ka * BS;         // K row ka
      const float* br1 = br0 + BS;             // K row ka+1
#pragma unroll
      for (int g = 0; g < 4; ++g) {
        const int nc = g * 16 + l16;
        v2f b;
        b.x = br0[nc];
        b.y = br1[nc];
        acc[g] = __builtin_amdgcn_wmma_f32_16x16x4_f32(
            false, a, false, b, (short)0, acc[g], false, false);
      }
    }
    __syncthreads();
  }

  // epilogue
#pragma unroll
  for (int g = 0; g < 4; ++g) {
    const int nc = colb + g * 16 + l16;
    float bs = (b1 ? b1[nc] : 0.f) + (b2 ? b2[nc] : 0.f);
#pragma unroll
    for (int r = 0; r < 8; ++r) {
      int row = row0 + r + lhalf * 8;          // C/D layout: vgpr r, lane half
      if (row < M) {
        float v = acc[g][r] + bs;
        if (slope >= 0.f) v = v > 0.f ? v : v * slope;
        C[(long)row * ldc + nc] = v;
      }
    }
  }
}

// ---------------- edge stage 1: logits + per-dst running max ----------------
// one wave per (edge, head); lane-strided channel loop + shuffle reduction
__global__ void edge_logits_kernel(const float* __restrict__ xl,
                                   const float* __restrict__ xr,
                                   const int* __restrict__ src,
                                   const int* __restrict__ dst,
                                   const float* __restrict__ att,
                                   const float* __restrict__ ea,
                                   const float* __restrict__ We,
                                   float* __restrict__ logits,
                                   unsigned* __restrict__ mx,
                                   int E, int H, int C) {
  int gw   = blockIdx.x * (blockDim.x >> 5) + (threadIdx.x >> 5);
  int lane = threadIdx.x & 31;
  if (gw >= E * H) return;
  int e = gw / H, h = gw % H;
  int s = src[e], d = dst[e];
  int HC = H * C;
  const float* pl = xl + (long)s * HC + h * C;
  const float* pr = xr + (long)d * HC + h * C;
  const float* pa = att + h * C;
  float ea7[7];
  if (We) {
#pragma unroll
    for (int k = 0; k < 7; ++k) ea7[k] = ea[(long)e * 7 + k];
  }
  float acc = 0.f;
  for (int c = lane; c < C; c += 32) {
    float m = pl[c] + pr[c];
    if (We) {
      float s2 = 0.f;
#pragma unroll
      for (int k = 0; k < 7; ++k) s2 += ea7[k] * We[k * HC + h * C + c];
      m += s2;
    }
    m = m > 0.f ? m : 0.2f * m;                // GATv2 internal leaky-relu
    acc += m * pa[c];
  }
#pragma unroll
  for (int o = 16; o > 0; o >>= 1) acc += __shfl_xor(acc, o, 32);
  if (lane == 0) {
    logits[(long)e * H + h] = acc;
    atomicMax(mx + (long)d * H + h, f2ord(acc));
  }
}

// ---------------- edge stage 2: exp(logit - max) and denominator -------------
__global__ void edge_softmax_kernel(float* __restrict__ lg,
                                    const int* __restrict__ dst,
                                    const unsigned* __restrict__ mx,
                                    float* __restrict__ den, int E, int H) {
  int i = blockIdx.x * blockDim.x + threadIdx.x;
  if (i >= E * H) return;
  int e = i / H, h = i % H;
  int d = dst[e];
  float ex = expf(lg[i] - ord2f(mx[(long)d * H + h]));
  lg[i] = ex;
  atomicAdd(den + (long)d * H + h, ex);
}

// ---------------- edge stage 3: out[dst] += alpha * xl[src] ------------------
__global__ void edge_aggregate_kernel(const float* __restrict__ xl,
                                      const float* __restrict__ ex,
                                      const int* __restrict__ src,
                                      const int* __restrict__ dst,
                                      const float* __restrict__ den,
                                      float* __restrict__ out, int ldo, int col0,
                                      int E, int H, int C) {
  int e = blockIdx.x;
  int c = threadIdx.x;                          // blockDim == H*C
  int h = c / C;
  int s = src[e], d = dst[e];
  float alpha = ex[(long)e * H + h] / (den[(long)d * H + h] + 1e-16f);
  atomicAdd(out + (long)d * ldo + col0 + c, xl[(long)s * H * C + c] * alpha);
}

// ---------------- per-node bias (+ optional leaky-relu) ----------------------
__global__ void bias_act_kernel(float* __restrict__ X, int ld, int col0,
                                int rows, int cols, const float* __restrict__ b,
                                float slope) {
  long i = (long)blockIdx.x * blockDim.x + threadIdx.x;
  if (i >= (long)rows * cols) return;
  int r = (int)(i / cols), c = (int)(i % cols);
  float v = X[(long)r * ld + col0 + c] + b[c];
  if (slope >= 0.f) v = v > 0.f ? v : v * slope;
  X[(long)r * ld + col0 + c] = v;
}

// ---------------- batch-norm: column stats + apply ---------------------------
__global__ void bn_stats_kernel(const float* __restrict__ X, int ld, int col0,
                                int rows, float* __restrict__ mean,
                                float* __restrict__ rstd) {
  __shared__ float ssum[256], ssq[256];
  int c = blockIdx.x, t = threadIdx.x;
  float s = 0.f, q = 0.f;
  for (int r = t; r < rows; r += 256) {
    float v = X[(long)r * ld + col0 + c];
    s += v; q += v * v;
  }
  ssum[t] = s; ssq[t] = q;
  __syncthreads();
  for (int o = 128; o > 0; o >>= 1) {
    if (t < o) { ssum[t] += ssum[t + o]; ssq[t] += ssq[t + o]; }
    __syncthreads();
  }
  if (t == 0) {
    float mu  = ssum[0] / rows;
    float var = ssq[0] / rows - mu * mu;
    mean[c] = mu;
    rstd[c] = rsqrtf(var + 1e-5f);
  }
}

__global__ void bn_apply_kernel(const float* __restrict__ X, int ldx, int colx,
                                float* __restrict__ Y, int ldy, int coly,
                                int rows, int cols, const float* __restrict__ mean,
                                const float* __restrict__ rstd,
                                const float* __restrict__ g,
                                const float* __restrict__ b, float slope) {
  long i = (long)blockIdx.x * blockDim.x + threadIdx.x;
  if (i >= (long)rows * cols) return;
  int r = (int)(i / cols), c = (int)(i % cols);
  float v = (X[(long)r * ldx + colx + c] - mean[c]) * rstd[c] * g[c] + b[c];
  if (slope >= 0.f) v = v > 0.f ? v : v * slope;
  Y[(long)r * ldy + coly + c] = v;
}

// =============================== host side ==================================
extern "C" void kernel_launch(void* const* d_in, const int* in_sizes, int n_in,
                              void* d_out, int out_size, void* d_ws,
                              size_t ws_size, hipStream_t stream) {
  (void)in_sizes; (void)n_in; (void)out_size; (void)ws_size;

  const float* x_r  = (const float*)d_in[0];   // [20000,128]
  const float* x_s  = (const float*)d_in[1];   // [5000,64]
  const int*   eirr = (const int*)d_in[2];     // [2,320000] row0=src row1=dst
  const int*   eirs = (const int*)d_in[3];     // [2,200000] row0=rec(dst) row1=src
  const float* earr = (const float*)d_in[4];   // [320000,7]
  const float* ears = (const float*)d_in[5];   // [200000,7]
  auto P = [&](int i) { return (const float*)d_in[i]; };

  float* ws = (float*)d_ws;
  // workspace layout (floats)
  const size_t o_cat  = 0;                     // [20000,768] concat(rec, cross)
  const size_t o_hA   = o_cat  + (size_t)NR * 768;
  const size_t o_hB   = o_hA   + (size_t)NR * 256;
  const size_t o_xl   = o_hB   + (size_t)NR * 256;   // >= 5000*384 too
  const size_t o_xr   = o_xl   + (size_t)NR * 256;   // [20000,384] for cross
  const size_t o_lg   = o_xr   + (size_t)NR * 384;   // [E*H]
  const size_t o_mx   = o_lg   + (size_t)ERR * 4;
  const size_t o_den  = o_mx   + (size_t)NR * 4;
  const size_t o_scat = o_den  + (size_t)NR * 4;     // [5000,384] source concat
  const size_t o_st1  = o_scat + (size_t)NS * 384;
  const size_t o_st2  = o_st1  + (size_t)NS * 256;
  const size_t o_mean = o_st2  + (size_t)NS * 256;
  const size_t o_rstd = o_mean + 768;
  const size_t o_tmp  = o_rstd + 768;                // [20000,128] final linears

  float*    cat  = ws + o_cat;
  float*    hA   = ws + o_hA;
  float*    hB   = ws + o_hB;
  float*    xl   = ws + o_xl;
  float*    xr   = ws + o_xr;
  float*    lg   = ws + o_lg;
  unsigned* mxu  = (unsigned*)(ws + o_mx);
  float*    den  = ws + o_den;
  float*    scat = ws + o_scat;
  float*    st1  = ws + o_st1;
  float*    st2  = ws + o_st2;
  float*    mean = ws + o_mean;
  float*    rstd = ws + o_rstd;
  float*    tmp  = ws + o_tmp;
  float*    out  = (float*)d_out;

  auto gemm = [&](const float* A, int lda, const float* W, const float* b1,
                  const float* b2, float* C, int ldc, int M, int N, int K,
                  float slope) {
    dim3 g(N / 64, (M + 63) / 64);
    gemm_bias_act<<<g, dim3(128), 0, stream>>>(A, lda, W, b1, b2, C, ldc, M, N,
                                               K, slope);
  };
  auto zf = [&](float* p, int n) {
    fill_f32<<<(n + 255) / 256, 256, 0, stream>>>(p, 0.f, n);
  };

  zf(cat, NR * 768);  // all aggregation targets in cat start at zero

  // one full GATv2 conv on the receptor-receptor graph
  auto rr_conv = [&](const float* hin, int ldh, int K, int pi, int H, int C,
                     float* o, int ldo, int col0, bool zero_out, float slope) {
    const int HC = H * C;
    gemm(hin, ldh, P(pi + 1), P(pi + 5), nullptr, xl, HC, NR, HC, K, -1.f); // Wl+bl
    gemm(hin, ldh, P(pi + 2), P(pi + 6), nullptr, xr, HC, NR, HC, K, -1.f); // Wr+br
    fill_u32<<<(NR * H + 255) / 256, 256, 0, stream>>>(mxu, 0x007FFFFFu, NR * H);
    zf(den, NR * H);
    if (zero_out) zf(o, NR * HC);
    const int* src = eirr;
    const int* dst = eirr + ERR;
    edge_logits_kernel<<<(ERR * H + 7) / 8, 256, 0, stream>>>(
        xl, xr, src, dst, P(pi + 3), earr, P(pi + 0), lg, mxu, ERR, H, C);
    edge_softmax_kernel<<<(ERR * H + 255) / 256, 256, 0, stream>>>(
        lg, dst, mxu, den, ERR, H);
    edge_aggregate_kernel<<<ERR, HC, 0, stream>>>(xl, lg, src, dst, den, o, ldo,
                                                  col0, ERR, H, C);
    bias_act_kernel<<<((long)NR * HC + 255) / 256, 256, 0, stream>>>(
        o, ldo, col0, NR, HC, P(pi + 4), slope);
  };

  // ---- receptor paths (HOPS 1,2,3) -> cat cols [0,128,256] ----
  rr_conv(x_r, 128, 128, 21, 1, 128, cat, 768, 0,   false, -1.f);
  rr_conv(x_r, 128, 128, 28, 4, 64,  hA,  256, 0,   true,  0.01f);
  rr_conv(hA,  256, 256, 35, 1, 128, cat, 768, 128, false, -1.f);
  rr_conv(x_r, 128, 128, 42, 4, 64,  hA,  256, 0,   true,  0.01f);
  rr_conv(hA,  256, 256, 49, 4, 64,  hB,  256, 0,   true,  0.01f);
  rr_conv(hB,  256, 256, 56, 1, 128, cat, 768, 256, false, -1.f);

  // ---- source paths: self-loop graph => alpha==1 => pure GEMM chains ----
  gemm(x_s, 64,  P(65), P(69), P(68), scat + 0,   384, NS, 128, 64,  -1.f);
  gemm(x_s, 64,  P(71), P(75), P(74), st1,        256, NS, 256, 64,  0.01f);
  gemm(st1, 256, P(77), P(81), P(80), scat + 128, 384, NS, 128, 256, -1.f);
  gemm(x_s, 64,  P(83), P(87), P(86), st1,        256, NS, 256, 64,  0.01f);
  gemm(st1, 256, P(89), P(93), P(92), st2,        256, NS, 256, 256, 0.01f);
  gemm(st2, 256, P(95), P(99), P(98), scat + 256, 384, NS, 128, 256, -1.f);

  // ---- batch norms: receptor_norm (cat cols 0..383), source_norm (scat) ----
  bn_stats_kernel<<<384, 256, 0, stream>>>(cat, 768, 0, NR, mean, rstd);
  bn_apply_kernel<<<((long)NR * 384 + 255) / 256, 256, 0, stream>>>(
      cat, 768, 0, cat, 768, 0, NR, 384, mean, rstd, P(20), P(19), -1.f);
  bn_stats_kernel<<<384, 256, 0, stream>>>(scat, 384, 0, NS, mean, rstd);
  bn_apply_kernel<<<((long)NS * 384 + 255) / 256, 256, 0, stream>>>(
      scat, 384, 0, scat, 384, 0, NS, 384, mean, rstd, P(64), P(63), -1.f);

  // ---- cross conv source->receptor: H=1, C=384, edges rs reversed ----
  gemm(scat, 384, P(7), P(11), nullptr, xl, 384, NS, 384, 384, -1.f); // Wl+bl
  gemm(cat,  768, P(8), P(12), nullptr, xr, 384, NR, 384, 384, -1.f); // Wr+br
  fill_u32<<<(NR + 255) / 256, 256, 0, stream>>>(mxu, 0x007FFFFFu, NR);
  zf(den, NR);
  const int* dstx = eirs;          // receptor ids
  const int* srcx = eirs + ERS;    // source ids
  edge_logits_kernel<<<(ERS + 7) / 8, 256, 0, stream>>>(
      xl, xr, srcx, dstx, P(9), ears, P(6), lg, mxu, ERS, 1, 384);
  edge_softmax_kernel<<<(ERS + 255) / 256, 256, 0, stream>>>(lg, dstx, mxu, den,
                                                             ERS, 1);
  edge_aggregate_kernel<<<ERS, 384, 0, stream>>>(xl, lg, srcx, dstx, den, cat,
                                                 768, 384, ERS, 1, 384);
  bias_act_kernel<<<((long)NR * 384 + 255) / 256, 256, 0, stream>>>(
      cat, 768, 384, NR, 384, P(10), -1.f);

  // ---- final linears + BN + leaky-relu -> d_out ----
  gemm(cat, 768, P(17), P(18), nullptr, tmp, 128, NR, 128, 768, -1.f);
  bn_stats_kernel<<<128, 256, 0, stream>>>(tmp, 128, 0, NR, mean, rstd);
  bn_apply_kernel<<<((long)NR * 128 + 255) / 256, 256, 0, stream>>>(
      tmp, 128, 0, out, 128, 0, NR, 128, mean, rstd, P(14), P(13), 0.01f);

  gemm(scat, 384, P(101), P(102), nullptr, tmp, 128, NS, 128, 384, -1.f);
  bn_stats_kernel<<<128, 256, 0, stream>>>(tmp, 128, 0, NS, mean, rstd);
  bn_apply_kernel<<<((long)NS * 128 + 255) / 256, 256, 0, stream>>>(
      tmp, 128, 0, out + (long)NR * 128, 128, 0, NS, 128, mean, rstd, P(16),
      P(15), 0.01f);
}